// ExpanderGCNLayer_44744969290330
// MI455X (gfx1250) — compile-verified
//
#include <hip/hip_runtime.h>

#define N_NODES 100000
#define N_EDGES 1600000
#define DIMS 32
#define LDS_STRIDE 34   // DIMS + 2 pad: keeps 8B alignment, avoids bank ping-pong
#define BN_EPS 1e-5f

typedef float v2f __attribute__((ext_vector_type(2)));
typedef float v8f __attribute__((ext_vector_type(8)));

// ---------------------------------------------------------------------------
// Kernel 1: zero the aggregation buffer + BN stats buffer
// ---------------------------------------------------------------------------
__global__ void gcn_zero_kernel(float* __restrict__ ws, int n) {
    int i = blockIdx.x * blockDim.x + threadIdx.x;
    if (i < n) ws[i] = 0.0f;
}

// ---------------------------------------------------------------------------
// Kernel 2: edge scatter-add. One wave per edge, one lane per channel.
// src/dst loads are wave-uniform; data stays in L2 (feature+h = 25.6MB << 192MB)
// ---------------------------------------------------------------------------
__global__ void gcn_scatter_kernel(const float* __restrict__ feat,
                                   const int* __restrict__ src,
                                   const int* __restrict__ dst,
                                   float* __restrict__ h,
                                   int nEdges) {
    long long gid = (long long)blockIdx.x * blockDim.x + threadIdx.x;
    int e    = (int)(gid >> 5);
    int lane = (int)(gid & 31);
    if (e < nEdges) {
        int s = src[e];
        int d = dst[e];
        atomicAdd(&h[(long long)d * DIMS + lane],
                  feat[(long long)s * DIMS + lane]);
    }
}

// ---------------------------------------------------------------------------
// Kernel 3: fp32 WMMA GEMM  y = (h @ W + b) * snorm, in place over h,
// with fused BN-statistics accumulation (sum / sumsq per column).
// Each wave: 16 rows x 32 cols via 2 accumulators, K=32 as 8 chained
// V_WMMA_F32_16X16X4_F32 steps. W is held TRANSPOSED in LDS so each
// B fragment {W[k][c], W[k+1][c]} is one contiguous ds_load_b64.
// ---------------------------------------------------------------------------
__global__ void gcn_gemm_bn_kernel(const float* __restrict__ W,
                                   const float* __restrict__ bias,
                                   const float* __restrict__ snorm,
                                   float* __restrict__ h,      // in: agg, out: y
                                   float* __restrict__ stats,  // [64]: sum, sumsq
                                   int nTiles) {
    __shared__ float Wt[DIMS * LDS_STRIDE];   // Wt[c][k] = W[k][c]
    for (int i = threadIdx.x; i < DIMS * DIMS; i += blockDim.x) {
        int k = i >> 5;        // input-dim index (row of W)
        int c = i & 31;        // output-dim index (col of W)
        Wt[c * LDS_STRIDE + k] = W[i];
    }
    __syncthreads();

    int wave = threadIdx.x >> 5;
    int lane = threadIdx.x & 31;
    int tile = blockIdx.x * (blockDim.x >> 5) + wave;
    if (tile >= nTiles) return;  // wave-uniform: EXEC stays all-ones for WMMA

    int idx15 = lane & 15;            // A: row-in-tile (M); B: column (N)
    int koff  = (lane >> 4) << 1;     // lane-half K offset: 0 or 2
    int arow  = tile * 16 + idx15;

    v8f c0 = {};  // output cols  0..15
    v8f c1 = {};  // output cols 16..31

#pragma unroll
    for (int kk = 0; kk < 8; ++kk) {
        int kr = kk * 4 + koff;
        // A fragment: h[arow][kr], h[arow][kr+1]  (8B-aligned pair)
        v2f a = *(const v2f*)&h[(long long)arow * DIMS + kr];
        // B fragments: contiguous pairs from transposed-W LDS (ds_load_b64)
        v2f b0 = *(const v2f*)&Wt[idx15 * LDS_STRIDE + kr];
        v2f b1 = *(const v2f*)&Wt[(idx15 + 16) * LDS_STRIDE + kr];
        c0 = __builtin_amdgcn_wmma_f32_16x16x4_f32(false, a, false, b0,
                                                   (short)0, c0, false, false);
        c1 = __builtin_amdgcn_wmma_f32_16x16x4_f32(false, a, false, b1,
                                                   (short)0, c1, false, false);
    }

    // Epilogue: + bias, * snorm, write y in place, accumulate BN partials.
    float bias0 = bias[idx15];
    float bias1 = bias[idx15 + 16];
    int   mbase = (lane >> 4) * 8;    // C/D: lanes 16-31 hold M = v + 8
    float psum0 = 0.f, psq0 = 0.f, psum1 = 0.f, psq1 = 0.f;

#pragma unroll
    for (int v = 0; v < 8; ++v) {
        int r = tile * 16 + mbase + v;
        float s  = snorm[r];
        float y0 = (c0[v] + bias0) * s;
        float y1 = (c1[v] + bias1) * s;
        h[(long long)r * DIMS + idx15]      = y0;
        h[(long long)r * DIMS + idx15 + 16] = y1;
        psum0 += y0;  psq0 += y0 * y0;
        psum1 += y1;  psq1 += y1 * y1;
    }
    atomicAdd(&stats[idx15],           psum0);
    atomicAdd(&stats[32 + idx15],      psq0);
    atomicAdd(&stats[idx15 + 16],      psum1);
    atomicAdd(&stats[32 + idx15 + 16], psq1);
}

// ---------------------------------------------------------------------------
// Kernel 4: finalize BN (mean/var from sums), ReLU, residual. Streaming pass.
// ---------------------------------------------------------------------------
__global__ void gcn_finalize_kernel(const float* __restrict__ feat,
                                    const float* __restrict__ y,
                                    const float* __restrict__ stats,
                                    const float* __restrict__ gamma,
                                    const float* __restrict__ beta,
                                    float* __restrict__ out,
                                    int total) {
    int i = blockIdx.x * blockDim.x + threadIdx.x;
    if (i >= total) return;
    int col = i & (DIMS - 1);
    const float invN = 1.0f / (float)N_NODES;
    float mean = stats[col] * invN;
    float var  = stats[32 + col] * invN - mean * mean;
    float rstd = rsqrtf(var + BN_EPS);
    float t = (y[i] - mean) * rstd * gamma[col] + beta[col];
    out[i] = feat[i] + fmaxf(t, 0.0f);
}

// ---------------------------------------------------------------------------
extern "C" void kernel_launch(void* const* d_in, const int* in_sizes, int n_in,
                              void* d_out, int out_size, void* d_ws, size_t ws_size,
                              hipStream_t stream) {
    const float* feat  = (const float*)d_in[0];
    const float* snorm = (const float*)d_in[1];
    const float* W     = (const float*)d_in[2];
    const float* b     = (const float*)d_in[3];
    const float* gamma = (const float*)d_in[4];
    const float* beta  = (const float*)d_in[5];
    const int*   src   = (const int*)d_in[6];
    const int*   dst   = (const int*)d_in[7];
    float*       out   = (float*)d_out;

    float* h     = (float*)d_ws;                        // N*D aggregation / y buffer
    float* stats = h + (size_t)N_NODES * DIMS;          // 64 floats

    // 1) zero accumulator + stats (deterministic per call)
    int zeroN = N_NODES * DIMS + 64;
    gcn_zero_kernel<<<(zeroN + 255) / 256, 256, 0, stream>>>(h, zeroN);

    // 2) scatter-add over edges: E waves, 32 lanes each
    long long sthreads = (long long)N_EDGES * 32;
    int sblocks = (int)((sthreads + 255) / 256);
    gcn_scatter_kernel<<<sblocks, 256, 0, stream>>>(feat, src, dst, h, N_EDGES);

    // 3) WMMA GEMM + graph-norm + BN statistics (8 waves/block, 16 rows/wave)
    int nTiles = N_NODES / 16;  // 6250
    gcn_gemm_bn_kernel<<<(nTiles + 7) / 8, 256, 0, stream>>>(W, b, snorm, h,
                                                             stats, nTiles);

    // 4) normalize + relu + residual
    int total = N_NODES * DIMS;
    gcn_finalize_kernel<<<(total + 255) / 256, 256, 0, stream>>>(
        feat, h, stats, gamma, beta, out, total);
}